// get_model_45535243272383
// MI455X (gfx1250) — compile-verified
//
#include <hip/hip_runtime.h>
#include <stdint.h>

// Farthest Point Sampling for MI455X (gfx1250).
// One persistent 1024-thread workgroup per batch (32 wave32 waves on one WGP).
// Per step: TDM (tensor_load_to_lds) streams the 3xN xyz planes tile-by-tile
// into double-buffered LDS while all waves update the running min-distance
// (kept in global scratch, L2-resident) and track a local argmax; then a
// two-level wave32 shuffle reduction picks the next centroid.

typedef unsigned int v4u __attribute__((ext_vector_type(4)));
typedef int          v8i __attribute__((ext_vector_type(8)));
typedef int          v4i __attribute__((ext_vector_type(4)));

#define BIGDIST   1e10f
#define FPS_BLOCK 1024
#define TILE_W    2048                 // points per DMA tile (x,y,z rows)
#define WAVES     (FPS_BLOCK / 32)     // 32 waves
#define PT_PER_TH (TILE_W / FPS_BLOCK) // 2 points per thread per tile

// Low 32 bits of a flat shared pointer == LDS byte offset (aperture mapping).
__device__ __forceinline__ uint32_t lds_off(const void* p) {
  return (uint32_t)(uintptr_t)p;
}

// Issue one TDM 2D tile load: rows y=0..2 (x,y,z planes, row stride N floats),
// cols x=0..TILE_W-1, into contiguous LDS [3][TILE_W] floats.
// D# layout per CDNA5 ISA ch.8 (group0 128b, group1 256b; groups 2/3 zero
// for a 2D tensor). This toolchain exposes the 6-arg builtin form:
// (v4u g0, v8i g1, v4i g2, v4i g3, v8i extra, i32 cpol).
__device__ __forceinline__ void tdm_load_tile(const float* tile_start, int N,
                                              uint32_t lds_byte) {
  uint64_t ga = (uint64_t)(uintptr_t)tile_start;
  v4u g0;
  g0.x = 1u;                                              // count=1, user D#
  g0.y = lds_byte;                                        // lds_addr
  g0.z = (uint32_t)(ga & 0xFFFFFFFFull);                  // global_addr[31:0]
  g0.w = (uint32_t)((ga >> 32) & 0x01FFFFFFull)           // global_addr[56:32]
       | (2u << 30);                                      // type=2 ("image")
  v8i g1;
  g1[0] = (int)(2u << 16);                                // data_size=2 (4B)
  g1[1] = (int)(((uint32_t)N & 0xFFFFu) << 16);           // tensor_dim0[15:0] @63:48
  g1[2] = (int)((((uint32_t)N >> 16) & 0xFFFFu)           // tensor_dim0[31:16]
       | (3u << 16));                                     // tensor_dim1=3
  g1[3] = (int)((uint32_t)TILE_W << 16);                  // tile_dim0 @127:112
  g1[4] = 3;                                              // tile_dim1=3, tile_dim2=0
  g1[5] = (int)(uint32_t)N;                               // tensor_dim0_stride[31:0]
  g1[6] = 0;                                              // stride hi / dim1_stride lo
  g1[7] = 0;
  v4i g2 = {0, 0, 0, 0};
  v4i g3 = {0, 0, 0, 0};
  v8i g4 = {0, 0, 0, 0, 0, 0, 0, 0};
  __builtin_amdgcn_tensor_load_to_lds(g0, g1, g2, g3, g4, 0);
}

__global__ __launch_bounds__(FPS_BLOCK)
void fps_kernel(const float* __restrict__ x, int N, int npoint,
                int* __restrict__ idx_out, float* __restrict__ samp_out,
                float* __restrict__ distws)
{
  __shared__ float tiles[2][3][TILE_W];   // double-buffered xyz tile (48 KB)
  __shared__ float s_val[WAVES];
  __shared__ int   s_idx[WAVES];
  __shared__ int   s_far;

  const int b    = (int)blockIdx.x;
  const int tid  = (int)threadIdx.x;
  const int lane = tid & 31;
  const int wave = tid >> 5;

  const float* __restrict__ X = x + (size_t)b * 3u * (size_t)N; // planar x,y,z
  float* __restrict__ dist = distws + (size_t)b * (size_t)N;
  const int ntiles = N / TILE_W;

  int farthest = 0;

  for (int t = 0; t < npoint; ++t) {
    // Centroid coords (uniform loads; L2-resident).
    const float cx = X[(size_t)farthest];
    const float cy = X[(size_t)N + farthest];
    const float cz = X[2u * (size_t)N + farthest];
    if (tid == 0) {
      idx_out[(size_t)b * npoint + t] = farthest;          // record PREVIOUS pick
      float* sp = samp_out + ((size_t)b * npoint + t) * 3;
      sp[0] = cx; sp[1] = cy; sp[2] = cz;
    }

    if (wave == 0)
      tdm_load_tile(X, N, lds_off(&tiles[0][0][0]));       // tile 0 -> buf 0

    float maxv = -1.0f;
    int   maxi = 0;

    for (int k = 0; k < ntiles; ++k) {
      if (wave == 0) {
        if (k + 1 < ntiles) {
          // Prefetch next tile into the other buffer, then wait for tile k
          // (TENSORcnt is in-order per wave: <=1 leaves only tile k+1 pending).
          tdm_load_tile(X + (size_t)(k + 1) * TILE_W, N,
                        lds_off(&tiles[(k + 1) & 1][0][0]));
          __builtin_amdgcn_s_wait_tensorcnt((short)1);
        } else {
          __builtin_amdgcn_s_wait_tensorcnt((short)0);
        }
      }
      __syncthreads();   // tile k visible in LDS to all waves

      const float* xx = &tiles[k & 1][0][0];
      const float* yy = &tiles[k & 1][1][0];
      const float* zz = &tiles[k & 1][2][0];

      const int j = tid * PT_PER_TH;         // fixed ownership: thread<->points
      const int n = k * TILE_W + j;

      float2 dv = *(const float2*)(dist + n);  // garbage at t==0, replaced by BIG

      float dx0 = xx[j]     - cx, dy0 = yy[j]     - cy, dz0 = zz[j]     - cz;
      float dx1 = xx[j + 1] - cx, dy1 = yy[j + 1] - cy, dz1 = zz[j + 1] - cz;
      float d0, d1;
      {
        // Match the reference's ((dx^2+dy^2)+dz^2) rounding: no fma contraction.
        #pragma clang fp contract(off)
        d0 = dx0 * dx0 + dy0 * dy0 + dz0 * dz0;
        d1 = dx1 * dx1 + dy1 * dy1 + dz1 * dz1;
      }

      float p0  = (t == 0) ? BIGDIST : dv.x;
      float p1  = (t == 0) ? BIGDIST : dv.y;
      float nd0 = fminf(p0, d0);
      float nd1 = fminf(p1, d1);
      *(float2*)(dist + n) = make_float2(nd0, nd1);

      if (nd0 > maxv) { maxv = nd0; maxi = n;     }  // strict > keeps lowest idx
      if (nd1 > maxv) { maxv = nd1; maxi = n + 1; }

      if (k + 1 < ntiles)  // gfx1250 global_prefetch_b8 for next dist tile
        __builtin_prefetch((const void*)(dist + n + TILE_W), 1, 0);

      __syncthreads();   // all waves done reading buf[k&1] before its reuse
    }

    // Wave32 shuffle argmax, lowest-index tie-break (== jnp.argmax semantics).
    #pragma unroll
    for (int off = 16; off > 0; off >>= 1) {
      float ov = __shfl_down(maxv, (unsigned)off, 32);
      int   oi = __shfl_down(maxi, (unsigned)off, 32);
      if (ov > maxv || (ov == maxv && oi < maxi)) { maxv = ov; maxi = oi; }
    }
    if (lane == 0) { s_val[wave] = maxv; s_idx[wave] = maxi; }
    __syncthreads();
    if (wave == 0) {       // WAVES == 32: one wave reduces all partials
      float v = s_val[lane];
      int   i = s_idx[lane];
      #pragma unroll
      for (int off = 16; off > 0; off >>= 1) {
        float ov = __shfl_down(v, (unsigned)off, 32);
        int   oi = __shfl_down(i, (unsigned)off, 32);
        if (ov > v || (ov == v && oi < i)) { v = ov; i = oi; }
      }
      if (lane == 0) s_far = i;
    }
    __syncthreads();
    farthest = s_far;
  }
}

extern "C" void kernel_launch(void* const* d_in, const int* in_sizes, int n_in,
                              void* d_out, int out_size, void* d_ws, size_t ws_size,
                              hipStream_t stream) {
  (void)n_in; (void)ws_size;
  const int B = 16;                                   // reference batch size
  const float* x = (const float*)d_in[0];             // (B, 3, N) planar f32
  const int N      = in_sizes[0] / (3 * B);           // 262144
  const int npoint = out_size / (4 * B);              // idx(B*np) + samp(B*np*3)

  int*   idx_out  = (int*)d_out;                      // first B*npoint ints
  float* samp_out = (float*)d_out + (size_t)B * npoint; // then B*npoint*3 floats
  float* distws   = (float*)d_ws;                     // B*N floats of scratch

  fps_kernel<<<dim3(B), dim3(FPS_BLOCK), 0, stream>>>(
      x, N, npoint, idx_out, samp_out, distws);
}